// ResidualDIVEQ_53017076302229
// MI455X (gfx1250) — compile-verified
//
#include <hip/hip_runtime.h>
#include <math.h>

#define N_ROWS 131072
#define DIMS 64
#define KCODES 1024
#define CBOOKS 4

typedef __attribute__((ext_vector_type(2))) float v2f;
typedef __attribute__((ext_vector_type(8))) float v8f;

__global__ void __launch_bounds__(256) zero_counts_kernel(unsigned* counts) {
    int i = blockIdx.x * 256 + threadIdx.x;
    if (i < CBOOKS * KCODES) counts[i] = 0u;
}

// One block = 256 threads = 8 waves; each wave owns 16 rows, kept in registers
// in the WMMA A-operand layout across all 4 residual-VQ stages.
// LDS: per-stage codebook (256KB) pre-swizzled to B-operand layout + code norms.
// Inner loop double-buffers the 16-code B tile in registers so DS latency
// overlaps the 16-deep v_wmma_f32_16x16x4_f32 accumulation chain.
__global__ void __launch_bounds__(256) rvq_main_kernel(
    const float* __restrict__ z, const float* __restrict__ noise,
    const float* __restrict__ cb, float* __restrict__ out,
    unsigned* __restrict__ counts) {
    extern __shared__ float lds[];
    v2f*   ldsB  = (v2f*)lds;        // [64 tiles][16 groups][32 lanes] float2 = 256KB
    float* cnorm = lds + 65536;      // [1024]

    const int tid   = threadIdx.x;
    const int lane  = tid & 31;
    const int wave  = tid >> 5;
    const int half  = lane >> 4;     // 0: K={0,1}, 1: K={2,3} (A layout)
    const int lraw  = lane & 15;
    const int kbase = 2 * half;
    const int rowBase = blockIdx.x * 128 + wave * 16;
    const int myRow   = rowBase + lraw;

    // residual rows in A-operand layout: a[g] = rem[myRow][4g+kbase .. +1]
    v2f a[16];
    const float* zr = z + (size_t)myRow * DIMS + kbase;
#pragma unroll
    for (int g = 0; g < 16; ++g) a[g] = *(const v2f*)(zr + 4 * g);

    float best[8];
    int   bidx[8];

    auto loadB = [&](int t, v2f (&bv)[16]) {
#pragma unroll
        for (int g = 0; g < 16; ++g) bv[g] = ldsB[(t * 16 + g) * 32 + lane];
    };
    auto mmaTile = [&](const v2f (&bv)[16]) -> v8f {
        v8f acc = {0.f, 0.f, 0.f, 0.f, 0.f, 0.f, 0.f, 0.f};
#pragma unroll
        for (int g = 0; g < 16; ++g)
            acc = __builtin_amdgcn_wmma_f32_16x16x4_f32(
                false, a[g], false, bv[g], (short)0, acc, false, false);
        return acc;
    };
    auto scoreUpdate = [&](int t, v8f acc) {
        float cn = cnorm[t * 16 + lraw];
        int col = t * 16 + lraw;
#pragma unroll
        for (int v = 0; v < 8; ++v) {
            float s = cn - 2.f * acc[v];  // ||c||^2 - 2 x.c (row term constant)
            if (s < best[v]) { best[v] = s; bidx[v] = col; }
        }
    };

    for (int c = 0; c < CBOOKS; ++c) {
        const float* cbc = cb + (size_t)c * KCODES * DIMS;
        // Fill LDS with stage codebook in B-operand layout:
        // element i = ((t*16+g)*32+ln) holds cb[t*16+(ln&15)][4g+2*(ln>>4) .. +1]
        for (int i = tid; i < 64 * 16 * 32; i += 256) {
            int t = i >> 9;
            int g = (i >> 5) & 15;
            int ln = i & 31;
            int code = t * 16 + (ln & 15);
            int dim = 4 * g + 2 * (ln >> 4);
            ldsB[i] = *(const v2f*)(cbc + code * DIMS + dim);
        }
        for (int k = tid; k < KCODES; k += 256) {
            const float* row = cbc + k * DIMS;
            float s = 0.f;
#pragma unroll
            for (int d = 0; d < DIMS; ++d) s += row[d] * row[d];
            cnorm[k] = s;
        }
        __syncthreads();

#pragma unroll
        for (int v = 0; v < 8; ++v) { best[v] = 3.4e38f; bidx[v] = 0; }

        // Double-buffered tile loop: prefetch B(t+1) while WMMA-ing B(t).
        v2f b0[16], b1[16];
        loadB(0, b0);
        for (int t = 0; t < 64; t += 2) {
            loadB(t + 1, b1);
            scoreUpdate(t, mmaTile(b0));
            if (t + 2 < 64) loadB(t + 2, b0);
            scoreUpdate(t + 1, mmaTile(b1));
        }

        // reduce across the 16 lanes of each half (rows 0..7 / 8..15)
#pragma unroll
        for (int off = 1; off < 16; off <<= 1) {
#pragma unroll
            for (int v = 0; v < 8; ++v) {
                float ob = __shfl_xor(best[v], off, 32);
                int   oi = __shfl_xor(bidx[v], off, 32);
                if (ob < best[v] || (ob == best[v] && oi < bidx[v])) {
                    best[v] = ob; bidx[v] = oi;
                }
            }
        }
        // broadcast row (lane&15)'s argmin to this lane (A-layout lanes)
        int srcLane = (lraw >> 3) << 4;
        int tv[8];
#pragma unroll
        for (int v = 0; v < 8; ++v) tv[v] = __shfl(bidx[v], srcLane, 32);
        int sel = lane & 7;
        int myIdx = tv[0];
#pragma unroll
        for (int v = 1; v < 8; ++v) myIdx = (sel == v) ? tv[v] : myIdx;

        if (lane < 16) {
            out[(size_t)N_ROWS * DIMS + (size_t)c * N_ROWS + rowBase + lane] =
                (float)myIdx;
            atomicAdd(&counts[c * KCODES + myIdx], 1u);
        }
        // rem -= cb[myIdx]; gather from swizzled LDS layout
        int tq = myIdx >> 4, cq = myIdx & 15;
#pragma unroll
        for (int g = 0; g < 16; ++g) {
            v2f q = ldsB[(tq * 16 + g) * 32 + cq + 16 * half];
            a[g].x -= q.x;
            a[g].y -= q.y;
        }
        __syncthreads();  // protect LDS before next stage overwrites it
    }

    // Epilogue: direction = -rem; rv = noise + direction; z_q = z + |dir| * rv/|rv|
    float s_rem = 0.f, s_rv = 0.f;
    const float* nr = noise + (size_t)myRow * DIMS + kbase;
#pragma unroll
    for (int g = 0; g < 16; ++g) {
        v2f nv = *(const v2f*)(nr + 4 * g);
        v2f rv;
        rv.x = nv.x - a[g].x;
        rv.y = nv.y - a[g].y;
        s_rem += a[g].x * a[g].x + a[g].y * a[g].y;
        s_rv  += rv.x * rv.x + rv.y * rv.y;
        a[g] = rv;
    }
    s_rem += __shfl_xor(s_rem, 16, 32);  // combine the two half-rows
    s_rv  += __shfl_xor(s_rv, 16, 32);
    float errm  = sqrtf(s_rem);
    float scale = errm / fmaxf(sqrtf(s_rv), 1e-12f);
    float* outr = out + (size_t)myRow * DIMS + kbase;
#pragma unroll
    for (int g = 0; g < 16; ++g) {
        v2f zv = *(const v2f*)(zr + 4 * g);
        v2f o;
        o.x = zv.x + scale * a[g].x;
        o.y = zv.y + scale * a[g].y;
        *(v2f*)(outr + 4 * g) = o;
    }
}

__global__ void __launch_bounds__(256) perplexity_kernel(
    const unsigned* __restrict__ counts, float* __restrict__ out) {
    __shared__ float red[256];
    int c = blockIdx.x;
    float s = 0.f;
    for (int k = threadIdx.x; k < KCODES; k += 256) {
        float p = (float)counts[c * KCODES + k] * (1.0f / (float)N_ROWS);
        s += p * logf(p + 1e-10f);
    }
    red[threadIdx.x] = s;
    __syncthreads();
    for (int o = 128; o > 0; o >>= 1) {
        if (threadIdx.x < o) red[threadIdx.x] += red[threadIdx.x + o];
        __syncthreads();
    }
    if (threadIdx.x == 0)
        out[(size_t)N_ROWS * DIMS + (size_t)CBOOKS * N_ROWS + c] = expf(-red[0]);
}

extern "C" void kernel_launch(void* const* d_in, const int* in_sizes, int n_in,
                              void* d_out, int out_size, void* d_ws, size_t ws_size,
                              hipStream_t stream) {
    (void)in_sizes; (void)n_in; (void)out_size; (void)ws_size;
    const float* z     = (const float*)d_in[0];
    const float* noise = (const float*)d_in[1];
    const float* cb    = (const float*)d_in[2];
    float*    out    = (float*)d_out;
    unsigned* counts = (unsigned*)d_ws;  // 4*1024 u32

    const size_t ldsBytes = (size_t)(65536 + 1024) * sizeof(float);  // 260KB < 320KB
    static int attrDone = 0;  // idempotent host-side attribute set (not a stream op)
    if (!attrDone) {
        hipFuncSetAttribute((const void*)rvq_main_kernel,
                            hipFuncAttributeMaxDynamicSharedMemorySize,
                            (int)ldsBytes);
        attrDone = 1;
    }

    zero_counts_kernel<<<(CBOOKS * KCODES + 255) / 256, 256, 0, stream>>>(counts);
    rvq_main_kernel<<<N_ROWS / 128, 256, ldsBytes, stream>>>(z, noise, cb, out, counts);
    perplexity_kernel<<<CBOOKS, 256, 0, stream>>>(counts, out);
}